// KMEANS_67980742361659
// MI455X (gfx1250) — compile-verified
//
#include <hip/hip_runtime.h>
#include <hip/hip_bf16.h>

typedef __attribute__((ext_vector_type(2))) float v2f;
typedef __attribute__((ext_vector_type(8))) float v8f;

#define NROW 262144
#define KC 256
#define DDIM 64
#define ALPHA 0.1f
// LDS row stride (floats) for centers. 68 => lane l reads dword addr 68*l (+2 for
// the upper half-wave): bank pairs (4l,4l+1) vs (4l+2,4l+3) — all 32 lanes hit
// disjoint bank pairs, so the ds_load_2addr_b64 B-fragment reads are conflict-free.
#define ROW_PAD 68
#define ROWS_PER_BLOCK 128
#define THREADS 256

// ---------------- workspace init (graph-capture safe, no memset) ----------------
__global__ void kmeans_init_ws(float* lossAcc, unsigned* counts) {
    int t = threadIdx.x;
    if (t == 0) *lossAcc = 0.0f;
    if (t < 100) counts[t] = 0u;
}

// Per-column-tile epilogue on the WMMA accumulator layout.
// NOTE: no index tie-break here — within a lane's sweep, column indices are
// strictly increasing, so strict '<' keeps the first (lowest-index) minimum.
__device__ __forceinline__ void tile_epilogue(
    const v8f& acc, int col, const float* __restrict__ sC2, const float* x2m,
    float* swe, float* swd, float* dmin, int* nmin)
{
    const float c2v = sC2[col];
    #pragma unroll
    for (int i = 0; i < 8; ++i) {
        float d = x2m[i] + c2v - 2.0f * acc[i];
        d = fmaxf(d, 0.0f);
        // (1+d)^(-alpha) == exp2(-alpha*log2(1+d)); exactly 2 hardware trans ops.
        float we = __builtin_amdgcn_exp2f(-ALPHA * __builtin_amdgcn_logf(1.0f + d));
        swe[i] += we;
        swd[i] += we * d;
        if (d < dmin[i]) { dmin[i] = d; nmin[i] = col; }
    }
}

// ---------------- fused distance + loss + argmin kernel ----------------
__global__ __launch_bounds__(THREADS) void kmeans_main(
    const float* __restrict__ x, const int* __restrict__ y,
    const float* __restrict__ centers,
    float* __restrict__ lossAcc, unsigned* __restrict__ counts)
{
    extern __shared__ float smem[];
    float* sC    = smem;                  // [KC][ROW_PAD] centers
    float* sC2   = smem + KC * ROW_PAD;   // [KC] center norms
    float* sLoss = sC2 + KC;              // [1] block loss accumulator

    const int tid  = threadIdx.x;
    const int lane = tid & 31;
    const int wave = tid >> 5;
    const int l16  = lane & 15;
    const int half = lane >> 4;

    const int row_base = blockIdx.x * ROWS_PER_BLOCK + wave * 16;
    const int myrow    = row_base + l16;

    // Kick off HBM fetch of this wave's x rows while we stage centers in LDS
    // (global_prefetch_b8 on gfx1250).
    {
        const char* xrow = (const char*)(x + (size_t)myrow * DDIM);
        __builtin_prefetch(xrow, 0, 3);
        __builtin_prefetch(xrow + 128, 0, 3);
    }

    // cooperatively stage all 256 centers (64 KB) into LDS
    for (int idx = tid; idx < KC * DDIM; idx += THREADS) {
        int c = idx >> 6;
        int d = idx & 63;
        sC[c * ROW_PAD + d] = centers[idx];
    }
    if (tid == 0) *sLoss = 0.0f;
    __syncthreads();

    for (int c = tid; c < KC; c += THREADS) {
        float s = 0.f;
        #pragma unroll 8
        for (int d = 0; d < DDIM; ++d) { float v = sC[c * ROW_PAD + d]; s += v * v; }
        sC2[c] = s;
    }
    __syncthreads();

    // Preload A fragments for this wave's 16x64 tile of x.
    // f32 16x4 A layout: lane%16 = M, lane/16 selects K pair {0,1} vs {2,3}.
    v2f a[16];
    float x2p = 0.f;
    #pragma unroll
    for (int kt = 0; kt < 16; ++kt) {
        a[kt] = *(const v2f*)(x + (size_t)myrow * DDIM + kt * 4 + 2 * half);
        x2p += a[kt].x * a[kt].x + a[kt].y * a[kt].y;
    }
    // full ||x_row||^2 (each lane holds half the K range; partner is lane^16)
    float x2row = x2p + __shfl_xor(x2p, 16, 32);
    // x2 for the C-layout rows this lane owns: M = i + 8*half
    float x2m[8];
    #pragma unroll
    for (int i = 0; i < 8; ++i) x2m[i] = __shfl(x2row, i + 8 * half, 32);

    float swe[8], swd[8], dmin[8];
    int   nmin[8];
    #pragma unroll
    for (int i = 0; i < 8; ++i) { swe[i] = 0.f; swd[i] = 0.f; dmin[i] = 3.4e38f; nmin[i] = 0; }

    // Sweep 256 centers, TWO 16-wide column tiles per iteration: tile1's LDS
    // loads and tile0's epilogue are independent of tile0's WMMA chain, so the
    // scheduler overlaps ds_load latency and epilogue VALU with matrix work.
    for (int ct = 0; ct < KC / 32; ++ct) {
        const int cb0 = ct * 32;
        const int cb1 = ct * 32 + 16;
        const float* bp0 = sC + (cb0 + l16) * ROW_PAD + 2 * half;
        const float* bp1 = sC + (cb1 + l16) * ROW_PAD + 2 * half;

        v2f b0[16], b1[16];
        #pragma unroll
        for (int kt = 0; kt < 16; ++kt) b0[kt] = *(const v2f*)(bp0 + kt * 4);
        #pragma unroll
        for (int kt = 0; kt < 16; ++kt) b1[kt] = *(const v2f*)(bp1 + kt * 4);

        v8f acc0 = {};
        #pragma unroll
        for (int kt = 0; kt < 16; ++kt)
            acc0 = __builtin_amdgcn_wmma_f32_16x16x4_f32(
                false, a[kt], false, b0[kt], (short)0, acc0, false, false);

        v8f acc1 = {};
        #pragma unroll
        for (int kt = 0; kt < 16; ++kt)
            acc1 = __builtin_amdgcn_wmma_f32_16x16x4_f32(
                false, a[kt], false, b1[kt], (short)0, acc1, false, false);

        tile_epilogue(acc0, cb0 + l16, sC2, x2m, swe, swd, dmin, nmin);
        tile_epilogue(acc1, cb1 + l16, sC2, x2m, swe, swd, dmin, nmin);
    }

    // butterfly reduce over the 16-lane half (matches C-matrix half layout);
    // index tie-break applied here so cross-lane ties pick the first index
    #pragma unroll
    for (int off = 1; off <= 8; off <<= 1) {
        #pragma unroll
        for (int i = 0; i < 8; ++i) {
            swe[i] += __shfl_xor(swe[i], off, 32);
            swd[i] += __shfl_xor(swd[i], off, 32);
            float od = __shfl_xor(dmin[i], off, 32);
            int   on = __shfl_xor(nmin[i], off, 32);
            if (od < dmin[i] || (od == dmin[i] && on < nmin[i])) { dmin[i] = od; nmin[i] = on; }
        }
    }

    // lanes 0 and 16 hold full results for rows M=i and M=8+i respectively
    if (l16 == 0) {
        float lw = 0.f;
        #pragma unroll
        for (int i = 0; i < 8; ++i) {
            lw += swd[i] / swe[i];                  // row loss = sum(we*d)/sum(we)
            int yp = nmin[i];
            if (yp < 10) {
                int row = row_base + i + 8 * half;
                int lab = y[row];
                if (lab >= 0 && lab < 10)
                    atomicAdd(&counts[yp * 10 + lab], 1u);
            }
        }
        atomicAdd(sLoss, lw);
    }
    __syncthreads();
    if (tid == 0) atomicAdd(lossAcc, *sLoss);
}

// ---------------- greedy label matching + output ----------------
__global__ void kmeans_finalize(const float* lossAcc, const unsigned* counts, float* out) {
    if (threadIdx.x != 0 || blockIdx.x != 0) return;
    bool  used[10];
    for (int i = 0; i < 10; ++i) used[i] = false;
    float correct = 0.f;
    for (int i = 0; i < 10; ++i) {
        float bin[10]; float rowsum = 0.f;
        for (int c = 0; c < 10; ++c) { bin[c] = (float)counts[i * 10 + c]; rowsum += bin[c]; }
        bool has = rowsum > 0.f;
        int label = 0; float best = bin[0];
        for (int c = 1; c < 10; ++c) if (bin[c] > best) { best = bin[c]; label = c; }
        if (used[label]) {                       // one masked re-argmax, like the reference
            int l2 = 0; float b2 = used[0] ? 0.f : bin[0];
            for (int c = 1; c < 10; ++c) { float v = used[c] ? 0.f : bin[c]; if (v > b2) { b2 = v; l2 = c; } }
            label = l2;
        }
        if (has) { correct += bin[label]; used[label] = true; }
    }
    out[0] = *lossAcc;
    out[1] = correct / (float)NROW;
}

extern "C" void kernel_launch(void* const* d_in, const int* in_sizes, int n_in,
                              void* d_out, int out_size, void* d_ws, size_t ws_size,
                              hipStream_t stream) {
    const float* x       = (const float*)d_in[0];
    const int*   y       = (const int*)d_in[1];
    const float* centers = (const float*)d_in[2];
    float*       out     = (float*)d_out;

    float*    lossAcc = (float*)d_ws;
    unsigned* counts  = (unsigned*)d_ws + 1;

    kmeans_init_ws<<<1, 128, 0, stream>>>(lossAcc, counts);

    size_t lds = (size_t)(KC * ROW_PAD + KC + 1) * sizeof(float);
    kmeans_main<<<NROW / ROWS_PER_BLOCK, THREADS, lds, stream>>>(x, y, centers, lossAcc, counts);

    kmeans_finalize<<<1, 32, 0, stream>>>(lossAcc, counts, out);
}